// DifferentiableESKF_41918880808962
// MI455X (gfx1250) — compile-verified
//
#include <hip/hip_runtime.h>
#include <math.h>

namespace {

constexpr int NB    = 118;   // buses
constexpr int NS    = 236;   // state dim
constexpr int NM    = 238;   // meas dim
constexpr int NP    = 240;   // padded dim (15 * 16)
constexpr int NHIST = 5;
constexpr int DM    = 64;
constexpr int DH    = 256;   // 4*DM
constexpr int TSTEPS= 64;
constexpr int BSZ   = 32;
constexpr int NT    = 256;   // threads / block (8 wave32)
constexpr int NWAVES= NT / 32;
constexpr int NTIL  = NP / 16;
constexpr int CSP   = 128;   // padded row stride for cos/sin matrices

typedef __attribute__((ext_vector_type(2))) float v2f;
typedef __attribute__((ext_vector_type(8))) float v8f;

struct Params {
  const float* z_seq; const float* G;   const float* Bsus; const float* P0lv;
  const float* W_emb; const float* b_emb;
  const float* Wq;    const float* Wk;  const float* Wv;   const float* Wo;
  const float* W1;    const float* b1;  const float* W2;   const float* b2;
  const float* Wdx;   const float* bdx; const float* Wqd;  const float* bqd;
  const float* Wr1;   const float* br1; const float* Wr2;  const float* br2;
  float* out; float* ws;
};

__device__ __forceinline__ float softplus_f(float x) {
  return (x > 20.0f) ? x : log1pf(expf(x));
}

// D(240x240) = [optional C] (+/-) A * B over zero-padded 240x240 f32 matrices,
// using V_WMMA_F32_16X16X4_F32 tiles. One 16x16 output tile per wave at a time.
// Fragment layouts per CDNA5 ISA 7.12.2 (wave32):
//   A 16x4 : lanes 0-15 -> M=lane, K = k0+{0,1}; lanes 16-31 -> K = k0+{2,3}
//   B 4x16 : vgpr0 = row k0 (lanes 0-15) / row k0+2 (lanes 16-31); vgpr1 = k0+1 / k0+3
//   C/D    : vgpr r -> M = tm + r (lanes 0-15) | tm + 8 + r (lanes 16-31)
template<bool ATRANS, bool BTRANS, bool LOADC, bool NEGA>
__device__ __forceinline__ void gemm240(const float* __restrict__ A,
                                        const float* __restrict__ Bm,
                                        const float* __restrict__ C,
                                        float* __restrict__ D,
                                        int tid) {
  const int lane = tid & 31;
  const int wave = tid >> 5;
  const int hf   = lane >> 4;   // lane half: 0 or 1
  const int l16  = lane & 15;
  for (int tile = wave; tile < NTIL * NTIL; tile += NWAVES) {
    const int tm = (tile / NTIL) * 16;
    const int tn = (tile % NTIL) * 16;
    v8f acc{};
    if (LOADC) {
#pragma unroll
      for (int r = 0; r < 8; ++r)
        acc[r] = C[(tm + hf * 8 + r) * NP + tn + l16];
    }
#pragma unroll 4
    for (int k0 = 0; k0 < NP; k0 += 4) {
      const int ka = k0 + hf * 2;
      v2f afrag, bfrag;
      if (ATRANS) {
        afrag[0] = A[(ka + 0) * NP + tm + l16];
        afrag[1] = A[(ka + 1) * NP + tm + l16];
      } else {
        const float2 tv = *(const float2*)(A + (tm + l16) * NP + ka);
        afrag[0] = tv.x; afrag[1] = tv.y;
      }
      if (NEGA) { afrag[0] = -afrag[0]; afrag[1] = -afrag[1]; }
      if (BTRANS) {
        const float2 tv = *(const float2*)(Bm + (tn + l16) * NP + ka);
        bfrag[0] = tv.x; bfrag[1] = tv.y;
      } else {
        bfrag[0] = Bm[(ka + 0) * NP + tn + l16];
        bfrag[1] = Bm[(ka + 1) * NP + tn + l16];
      }
      acc = __builtin_amdgcn_wmma_f32_16x16x4_f32(false, afrag, false, bfrag,
                                                  (short)0, acc, false, false);
    }
#pragma unroll
    for (int r = 0; r < 8; ++r)
      D[(tm + hf * 8 + r) * NP + tn + l16] = acc[r];
  }
}

// Persistent kernel: one workgroup per batch element, full T-step scan inside.
__global__ __launch_bounds__(NT) void eskf_persistent(Params p) {
  const int tid = threadIdx.x;
  const int b   = blockIdx.x;

  // per-batch workspace (all L2-resident: ~1.3 MB/batch, 41 MB total)
  float* base = p.ws + (size_t)b * (5ull * NP * NP + 2ull * CSP * CSP);
  float* Pg  = base;                       // covariance 240x240
  float* Hg  = base + 1ull * NP * NP;      // Jacobian
  float* HPg = base + 2ull * NP * NP;      // H @ P
  float* Sg  = base + 3ull * NP * NP;      // innovation cov / Cholesky L
  float* Xg  = base + 4ull * NP * NP;      // S^{-1} HP
  float* CSC = base + 5ull * NP * NP;      // cos(va_i - va_j)
  float* CSS = CSC + CSP * CSP;            // sin(va_i - va_j)

  __shared__ float sh_x[NP];
  __shared__ float sh_zbuf[NHIST][NP];     // circular token buffer
  __shared__ float sh_e[NHIST][DM];
  __shared__ float sh_q[NHIST][DM];        // reused as att@v
  __shared__ float sh_k[NHIST][DM];
  __shared__ float sh_v[NHIST][DM];
  __shared__ float sh_att[NHIST][NHIST];
  __shared__ float sh_a[NHIST][DM];
  __shared__ float sh_hid[NHIST][DH];
  __shared__ float sh_pool[DM];
  __shared__ float sh_r1[DM];
  __shared__ float sh_dx[NP];
  __shared__ float sh_Qd[NP];
  __shared__ float sh_Rd[NP];
  __shared__ float sh_y[NP];
  __shared__ float sh_vm[CSP];
  __shared__ float sh_va[CSP];
  __shared__ float sh_t[CSP];              // sum_j (Gc+Bs) vm_j
  __shared__ float sh_u[CSP];              // sum_j (Gs-Bc) vm_j
  __shared__ float sh_diag;

  // ---- init: P0 = diag(exp(P0_logvar)), x0 = [1,0], zbuf = 0 ----
  for (int i = tid; i < NP * NP; i += NT) Pg[i] = 0.0f;
  for (int i = tid; i < NHIST * NP; i += NT) (&sh_zbuf[0][0])[i] = 0.0f;
  for (int s = tid; s < NP; s += NT) sh_x[s] = (s < NB) ? 1.0f : 0.0f;
  __syncthreads();
  for (int s = tid; s < NS; s += NT) Pg[s * NP + s] = expf(p.P0lv[s]);
  __syncthreads();

  for (int t = 0; t < TSTEPS; ++t) {
    // ---- append zk into circular slot (token order irrelevant: attn+mean pool
    //      are permutation invariant) ----
    float* zk = &sh_zbuf[t % NHIST][0];
    const float* zin = p.z_seq + ((size_t)b * TSTEPS + t) * NM;
    for (int j = tid; j < NP; j += NT) zk[j] = (j < NM) ? zin[j] : 0.0f;
    __syncthreads();

    // ---- error_net: e = zbuf @ W_emb + b_emb ----
    for (int o = tid; o < NHIST * DM; o += NT) {
      const int r = o / DM, c = o % DM;
      float acc = p.b_emb[c];
      for (int j = 0; j < NM; ++j) acc += sh_zbuf[r][j] * p.W_emb[j * DM + c];
      sh_e[r][c] = acc;
    }
    __syncthreads();
    // q,k,v
    for (int o = tid; o < NHIST * DM; o += NT) {
      const int r = o / DM, c = o % DM;
      float aq = 0.f, ak = 0.f, av = 0.f;
      for (int j = 0; j < DM; ++j) {
        const float ej = sh_e[r][j];
        aq += ej * p.Wq[j * DM + c];
        ak += ej * p.Wk[j * DM + c];
        av += ej * p.Wv[j * DM + c];
      }
      sh_q[r][c] = aq; sh_k[r][c] = ak; sh_v[r][c] = av;
    }
    __syncthreads();
    // attention scores (5x5), scale = 1/sqrt(64)
    if (tid < NHIST * NHIST) {
      const int r = tid / NHIST, c2 = tid % NHIST;
      float acc = 0.f;
      for (int j = 0; j < DM; ++j) acc += sh_q[r][j] * sh_k[c2][j];
      sh_att[r][c2] = acc * 0.125f;
    }
    __syncthreads();
    if (tid < NHIST) {
      float mx = sh_att[tid][0];
      for (int j2 = 1; j2 < NHIST; ++j2) mx = fmaxf(mx, sh_att[tid][j2]);
      float sm = 0.f;
      for (int j2 = 0; j2 < NHIST; ++j2) {
        const float e2 = expf(sh_att[tid][j2] - mx);
        sh_att[tid][j2] = e2; sm += e2;
      }
      const float inv = 1.0f / sm;
      for (int j2 = 0; j2 < NHIST; ++j2) sh_att[tid][j2] *= inv;
    }
    __syncthreads();
    // att @ v  (q buffer is dead -> reuse)
    for (int o = tid; o < NHIST * DM; o += NT) {
      const int r = o / DM, c = o % DM;
      float acc = 0.f;
      for (int j = 0; j < NHIST; ++j) acc += sh_att[r][j] * sh_v[j][c];
      sh_q[r][c] = acc;
    }
    __syncthreads();
    // a = (att@v) @ Wo + e
    for (int o = tid; o < NHIST * DM; o += NT) {
      const int r = o / DM, c = o % DM;
      float acc = sh_e[r][c];
      for (int j = 0; j < DM; ++j) acc += sh_q[r][j] * p.Wo[j * DM + c];
      sh_a[r][c] = acc;
    }
    __syncthreads();
    // hid = relu(a @ W1 + b1)
    for (int o = tid; o < NHIST * DH; o += NT) {
      const int r = o / DH, c = o % DH;
      float acc = p.b1[c];
      for (int j = 0; j < DM; ++j) acc += sh_a[r][j] * p.W1[j * DH + c];
      sh_hid[r][c] = fmaxf(acc, 0.0f);
    }
    __syncthreads();
    // f = hid @ W2 + b2 + a   (e buffer dead -> reuse)
    for (int o = tid; o < NHIST * DM; o += NT) {
      const int r = o / DM, c = o % DM;
      float acc = p.b2[c] + sh_a[r][c];
      for (int j = 0; j < DH; ++j) acc += sh_hid[r][j] * p.W2[j * DM + c];
      sh_e[r][c] = acc;
    }
    __syncthreads();
    // pooled = mean over tokens
    if (tid < DM) {
      float acc = 0.f;
      for (int r = 0; r < NHIST; ++r) acc += sh_e[r][tid];
      sh_pool[tid] = acc * (1.0f / NHIST);
    }
    __syncthreads();
    // dx, Qd ; r1 of meas-noise net (independent, same stage)
    for (int s = tid; s < NS; s += NT) {
      float adx = p.bdx[s], aqd = p.bqd[s];
      for (int j = 0; j < DM; ++j) {
        const float pj = sh_pool[j];
        adx += pj * p.Wdx[j * NS + s];
        aqd += pj * p.Wqd[j * NS + s];
      }
      sh_dx[s] = adx;
      sh_Qd[s] = softplus_f(aqd) + 1e-6f;
    }
    if (tid < DM) {
      float acc = p.br1[tid];
      for (int j = 0; j < NM; ++j) acc += zk[j] * p.Wr1[j * DM + tid];
      sh_r1[tid] = fmaxf(acc, 0.0f);
    }
    __syncthreads();
    // Rd ; x_pred = x + dx ; P_pred = P + diag(Qd)
    for (int m = tid; m < NM; m += NT) {
      float acc = p.br2[m];
      for (int j = 0; j < DM; ++j) acc += sh_r1[j] * p.Wr2[j * NM + m];
      sh_Rd[m] = softplus_f(acc) + 1e-4f;
    }
    if (tid < NS) {
      sh_x[tid] += sh_dx[tid];
      Pg[tid * NP + tid] += sh_Qd[tid];
    }
    __syncthreads();
    // vm / va
    if (tid < NB) { sh_vm[tid] = sh_x[tid]; sh_va[tid] = sh_x[NB + tid]; }
    __syncthreads();
    // cos/sin(va_i - va_j)
    for (int idx = tid; idx < NB * NB; idx += NT) {
      const int i = idx / NB, j = idx % NB;
      float sn, cc;
      sincosf(sh_va[i] - sh_va[j], &sn, &cc);
      CSC[i * CSP + j] = cc;
      CSS[i * CSP + j] = sn;
    }
    __syncthreads();
    // t_i, u_i
    if (tid < NB) {
      const int i = tid;
      float ta = 0.f, ua = 0.f;
      for (int j = 0; j < NB; ++j) {
        const float cc = CSC[i * CSP + j], sn = CSS[i * CSP + j];
        const float g = p.G[i * NB + j],  bb = p.Bsus[i * NB + j];
        const float vj = sh_vm[j];
        ta += (g * cc + bb * sn) * vj;
        ua += (g * sn - bb * cc) * vj;
      }
      sh_t[i] = ta; sh_u[i] = ua;
    }
    __syncthreads();
    // y = zk - h(x_pred)   and analytic Jacobian H (zero-padded 240x240)
    for (int m = tid; m < NP; m += NT) {
      float zp = 0.0f;
      if (m < NB)            zp = sh_vm[m] * sh_t[m];
      else if (m < NS)       zp = sh_vm[m - NB] * sh_u[m - NB];
      else if (m == NS)      zp = sh_vm[0];
      else if (m == NS + 1)  zp = sh_va[0];
      sh_y[m] = (m < NM) ? (zk[m] - zp) : 0.0f;
    }
    for (int idx = tid; idx < NP * NP; idx += NT) {
      const int m = idx / NP, s = idx % NP;
      float val = 0.0f;
      if (m < NS && s < NS) {
        const bool prow = (m < NB);
        const int  i = prow ? m : m - NB;
        const float vmi = sh_vm[i];
        if (s < NB) {                       // d/dvm_k
          const int kk = s;
          const float cc = CSC[i * CSP + kk], sn = CSS[i * CSP + kk];
          const float g  = p.G[i * NB + kk], bb = p.Bsus[i * NB + kk];
          if (prow) val = (kk == i) ? (sh_t[i] + vmi * g)
                                    : vmi * (g * cc + bb * sn);
          else      val = (kk == i) ? (sh_u[i] - vmi * bb)
                                    : vmi * (g * sn - bb * cc);
        } else {                            // d/dva_k
          const int kk = s - NB;
          const float cc = CSC[i * CSP + kk], sn = CSS[i * CSP + kk];
          const float g  = p.G[i * NB + kk], bb = p.Bsus[i * NB + kk];
          if (prow) val = (kk == i) ? (-vmi * sh_u[i] - bb * vmi * vmi)
                                    : vmi * sh_vm[kk] * (g * sn - bb * cc);
          else      val = (kk == i) ? (vmi * sh_t[i] - g * vmi * vmi)
                                    : -vmi * sh_vm[kk] * (g * cc + bb * sn);
        }
      } else if (m == NS && s == 0)      val = 1.0f;   // d vm0 / d vm0
      else if (m == NS + 1 && s == NB)   val = 1.0f;   // d va0 / d va0
      Hg[idx] = val;
    }
    __syncthreads();

    // ---- heavy linear algebra on WMMA f32 path ----
    gemm240<false, false, false, false>(Hg, Pg, nullptr, HPg, tid);   // HP = H P
    __syncthreads();
    gemm240<false, true,  false, false>(HPg, Hg, nullptr, Sg, tid);   // S = HP H^T
    __syncthreads();
    if (tid < NM) Sg[tid * NP + tid] += sh_Rd[tid];
    __syncthreads();

    // Cholesky S = L L^T (lower, in place, 238x238)
    for (int j = 0; j < NM; ++j) {
      if (tid == 0) {
        const float d2 = sqrtf(fmaxf(Sg[j * NP + j], 1e-20f));
        sh_diag = d2;
        Sg[j * NP + j] = d2;
      }
      __syncthreads();
      const float dinv = 1.0f / sh_diag;
      for (int i = j + 1 + tid; i < NM; i += NT) Sg[i * NP + j] *= dinv;
      __syncthreads();
      for (int i = j + 1 + tid; i < NM; i += NT) {
        const float lij = Sg[i * NP + j];
        for (int c = j + 1; c <= i; ++c) Sg[i * NP + c] -= lij * Sg[c * NP + j];
      }
      __syncthreads();
    }

    // X = S^{-1} HP : one RHS column per thread (L reads are lane-uniform,
    // X accesses coalesced). Pad columns/rows of X zeroed for the WMMA pass.
    if (tid < NS) {
      const int c = tid;
      for (int i = 0; i < NM; ++i) {
        float acc = HPg[i * NP + c];
        for (int k2 = 0; k2 < i; ++k2) acc -= Sg[i * NP + k2] * Xg[k2 * NP + c];
        Xg[i * NP + c] = acc / Sg[i * NP + i];
      }
      for (int i = NM - 1; i >= 0; --i) {
        float acc = Xg[i * NP + c];
        for (int k2 = i + 1; k2 < NM; ++k2) acc -= Sg[k2 * NP + i] * Xg[k2 * NP + c];
        Xg[i * NP + c] = acc / Sg[i * NP + i];
      }
    } else if (tid < NP) {
      for (int i = 0; i < NP; ++i) Xg[i * NP + tid] = 0.0f;
    }
    __syncthreads();
    if (tid < NP) { Xg[NM * NP + tid] = 0.0f; Xg[(NM + 1) * NP + tid] = 0.0f; }
    __syncthreads();

    // x_est = x_pred + K y = x_pred + X^T y ; emit output
    if (tid < NS) {
      float acc = sh_x[tid];
      for (int m = 0; m < NM; ++m) acc += Xg[m * NP + tid] * sh_y[m];
      sh_x[tid] = acc;
      p.out[((size_t)b * TSTEPS + t) * NS + tid] = acc;
    }
    __syncthreads();

    // P_est = P_pred - X^T HP  (A negated at load, C preloaded from P, in place)
    gemm240<true, false, true, true>(Xg, HPg, Pg, Pg, tid);
    __syncthreads();
  }
}

} // namespace

extern "C" void kernel_launch(void* const* d_in, const int* in_sizes, int n_in,
                              void* d_out, int out_size, void* d_ws, size_t ws_size,
                              hipStream_t stream) {
  (void)in_sizes; (void)n_in; (void)out_size; (void)ws_size;
  Params p;
  p.z_seq = (const float*)d_in[0];
  p.G     = (const float*)d_in[1];
  p.Bsus  = (const float*)d_in[2];
  p.P0lv  = (const float*)d_in[3];
  p.W_emb = (const float*)d_in[4];
  p.b_emb = (const float*)d_in[5];
  p.Wq    = (const float*)d_in[6];
  p.Wk    = (const float*)d_in[7];
  p.Wv    = (const float*)d_in[8];
  p.Wo    = (const float*)d_in[9];
  p.W1    = (const float*)d_in[10];
  p.b1    = (const float*)d_in[11];
  p.W2    = (const float*)d_in[12];
  p.b2    = (const float*)d_in[13];
  p.Wdx   = (const float*)d_in[14];
  p.bdx   = (const float*)d_in[15];
  p.Wqd   = (const float*)d_in[16];
  p.bqd   = (const float*)d_in[17];
  p.Wr1   = (const float*)d_in[18];
  p.br1   = (const float*)d_in[19];
  p.Wr2   = (const float*)d_in[20];
  p.br2   = (const float*)d_in[21];
  p.out   = (float*)d_out;
  p.ws    = (float*)d_ws;
  eskf_persistent<<<dim3(BSZ), dim3(NT), 0, stream>>>(p);
}